// RotamerProjection_80736795230625
// MI455X (gfx1250) — compile-verified
//
#include <hip/hip_runtime.h>

// ---------------------------------------------------------------------------
// RotamerProjection for MI455X (gfx1250, wave32).
//
// All 4x4 frame compositions are batched onto V_WMMA_F32_16X16X4_F32.
// One WMMA = 4 independent 4x4 products (diagonal blocks of the 16x16 result):
//   A (16x4)  = 4 residues' left matrices stacked vertically (rows 4i..4i+3)
//   B (4x16)  = 4 residues' right matrices side by side      (cols 4i..4i+3)
//   D block(i,i) = L_i @ R_i
// Each wave owns 4 residues + a private 4KB LDS region:
//   floats [0,512)   : [residue i][frame a][4x4] frame buffer
//   floats [512,1024): dump area for the non-diagonal D elements
//
// All LDS addresses are (lane-constant base pointer) + (compile-time slot
// offset) => ds ops carry immediate offsets, no per-iteration address VALU.
// All global table/dihedral loads are (hoisted base) + immediate offset and
// are issued up front for memory-level parallelism.
//
// Phases: (1) fr[a] = default_frames[seq][a] @ rot(sin,cos)   -> 8 WMMA
//         (2) chi chain fr[f] = fr[f-1] @ fr[f], f=5..7       -> 3 WMMA
//         (3) gfr[a] = bb_mat @ fr[a]                         -> 8 WMMA
//         (4) per-atom gather + affine transform + mask (VALU)
// ---------------------------------------------------------------------------

typedef float v2f __attribute__((ext_vector_type(2)));
typedef float v8f __attribute__((ext_vector_type(8)));

#define RES_TOTAL   (32 * 4096)
#define RES_PER_BLK 32               // 8 waves * 4 residues
#define WAVE_LDS    1024             // floats: 512 frames + 512 dump
#define EPSF        1e-12f

__device__ __forceinline__ void wave_sync() {
  __builtin_amdgcn_wave_barrier();
}

__device__ __forceinline__ v8f wmma4x4(v2f a, v2f b) {
  v8f c = {};
  // 8-arg form: (neg_a, A, neg_b, B, c_mod, C, reuse_a, reuse_b)
  return __builtin_amdgcn_wmma_f32_16x16x4_f32(false, a, false, b, (short)0, c,
                                               false, false);
}

__global__ __launch_bounds__(256) void rotamer_projection_kernel(
    const float* __restrict__ dih,      // (B,R,7,2)
    const float* __restrict__ bbc,      // (B,R,3,3)
    const int*   __restrict__ seq,      // (B,R)
    const float* __restrict__ dframes,  // (21,8,4,4)
    const int*   __restrict__ gidx,     // (21,14)
    const float* __restrict__ amask,    // (21,14)
    const float* __restrict__ litp,     // (21,14,3)
    float*       __restrict__ out)      // (B,R,14,3)
{
  __shared__ float ldsfr[8][WAVE_LDS];

  const int tid  = threadIdx.x;
  const int wave = tid >> 5;
  const int lane = tid & 31;
  float* fr = ldsfr[wave];

  const int low = lane & 15;   // A-row / B-col / D-col index
  const int hi  = lane >> 4;   // K-half select
  const int i   = low >> 2;    // residue slot 0..3 within the wave
  const int m   = low & 3;     // row-in-block (A) == col-in-block (B/D)
  const int k0  = hi * 2;      // first K index held by this lane

  const int resBase = blockIdx.x * RES_PER_BLK + wave * 4;
  const int myRes   = resBase + i;
  const int mySeq   = seq[myRes];

  // Lane-constant D-store pointers: accumulator element v holds D row
  // (v + 8*hi); kept iff that row's 4x4 block equals i. Kept elements land at
  // fr[i*128 + slot*16 + r*4 + m], others in the dump half. slot*16 is a
  // compile-time immediate at every use site.
  float* stq[8];
#pragma unroll
  for (int v = 0; v < 8; ++v) {
    const int row  = v + 8 * hi;
    const bool keep = (row >> 2) == i;
    const int off  = keep ? ((row & 3) * 4 + m) : (512 + low);
    stq[v] = fr + i * 128 + off;
  }
  // Lane-constant operand base pointers (frame slot = immediate offset).
  float* pA = fr + i * 128 + m * 4 + k0;   // A: row (i,m), cols k0..k0+1
  float* pB = fr + i * 128 + k0 * 4 + m;   // B: rows k0,k0+1, col (i,m)

  // Lane-constant weights for the rotation-matrix B elements:
  // rot = [[1,0,0,0],[0,c,-s,0],[0,s,c,0],[0,0,0,1]];  elem = wc*c + ws*s + wk
  float wc[2], ws[2], wk[2];
#pragma unroll
  for (int e = 0; e < 2; ++e) {
    const int k = k0 + e;
    wc[e] = ((k == 1 && m == 1) || (k == 2 && m == 2)) ? 1.f : 0.f;
    ws[e] = (k == 2 && m == 1) ? 1.f : ((k == 1 && m == 2) ? -1.f : 0.f);
    wk[e] = ((k == 0 && m == 0) || (k == 3 && m == 3)) ? 1.f : 0.f;
  }

  // ---- Up-front global loads (hoisted bases, immediate offsets) ----------
  // Default-frame A operands: 8 x b64.
  const float* dfBase = dframes + mySeq * 128 + m * 4 + k0;
  v2f dfv[8];
#pragma unroll
  for (int a = 0; a < 8; ++a) {
    dfv[a][0] = dfBase[a * 16];
    dfv[a][1] = dfBase[a * 16 + 1];
  }
  // Dihedrals: 7 x b64, then normalize to (sin, cos).
  const float* dBase = dih + myRes * 14;
  float sv[8], cv[8];
  sv[0] = 0.f;
  cv[0] = 1.f;
#pragma unroll
  for (int a = 1; a < 8; ++a) {
    const float x = dBase[(a - 1) * 2];
    const float y = dBase[(a - 1) * 2 + 1];
    const float inv = rsqrtf(x * x + y * y + EPSF);
    sv[a] = x * inv;
    cv[a] = y * inv;
  }

  // ---- Phase 1: fr[a] = default_frames[seq][a] @ rot_a -------------------
#pragma unroll
  for (int a = 0; a < 8; ++a) {
    v2f bop;
    bop[0] = fmaf(wc[0], cv[a], fmaf(ws[0], sv[a], wk[0]));
    bop[1] = fmaf(wc[1], cv[a], fmaf(ws[1], sv[a], wk[1]));

    v8f d = wmma4x4(dfv[a], bop);
#pragma unroll
    for (int v = 0; v < 8; ++v) stq[v][a * 16] = d[v];
    wave_sync();
  }

  // ---- Phase 2: chi chain, fr[f] <- fr[f-1] @ fr[f], f = 5..7 ------------
#pragma unroll
  for (int f = 5; f <= 7; ++f) {
    v2f aop;
    aop[0] = pA[(f - 1) * 16];
    aop[1] = pA[(f - 1) * 16 + 1];
    v2f bop;
    bop[0] = pB[f * 16];
    bop[1] = pB[f * 16 + 4];
    wave_sync();  // all lanes' reads of fr[f] precede the in-place overwrite

    v8f d = wmma4x4(aop, bop);
#pragma unroll
    for (int v = 0; v < 8; ++v) stq[v][f * 16] = d[v];
    wave_sync();
  }

  // ---- Backbone rigid frame (redundant per lane, cheap) ------------------
  const float* bp = bbc + myRes * 9;
  const float nx = bp[0], ny = bp[1], nz = bp[2];
  const float cax = bp[3], cay = bp[4], caz = bp[5];
  const float cx = bp[6], cy = bp[7], cz = bp[8];
  const float v1x = cx - cax, v1y = cy - cay, v1z = cz - caz;
  const float v2x = nx - cax, v2y = ny - cay, v2z = nz - caz;
  const float inv1 = rsqrtf(v1x * v1x + v1y * v1y + v1z * v1z + EPSF);
  const float e1x = v1x * inv1, e1y = v1y * inv1, e1z = v1z * inv1;
  const float d12 = e1x * v2x + e1y * v2y + e1z * v2z;
  const float u2x = v2x - e1x * d12, u2y = v2y - e1y * d12,
              u2z = v2z - e1z * d12;
  const float inv2 = rsqrtf(u2x * u2x + u2y * u2y + u2z * u2z + EPSF);
  const float e2x = u2x * inv2, e2y = u2y * inv2, e2z = u2z * inv2;
  const float e3x = e1y * e2z - e1z * e2y;
  const float e3y = e1z * e2x - e1x * e2z;
  const float e3z = e1x * e2y - e1y * e2x;

  // bb_mat row m, cols k0..k0+1 (cols = [e1 e2 e3 ca]; row 3 = 0 0 0 1)
  float row0, row1, row2, row3;
  if (m == 0)      { row0 = e1x; row1 = e2x; row2 = e3x; row3 = cax; }
  else if (m == 1) { row0 = e1y; row1 = e2y; row2 = e3y; row3 = cay; }
  else if (m == 2) { row0 = e1z; row1 = e2z; row2 = e3z; row3 = caz; }
  else             { row0 = 0.f; row1 = 0.f; row2 = 0.f; row3 = 1.f; }
  v2f abb;
  abb[0] = (k0 == 0) ? row0 : row2;
  abb[1] = (k0 == 0) ? row1 : row3;

  // ---- Phase 3: gfr[a] = bb_mat @ fr[a] (in place) -----------------------
#pragma unroll
  for (int a = 0; a < 8; ++a) {
    v2f bop;
    bop[0] = pB[a * 16];
    bop[1] = pB[a * 16 + 4];
    wave_sync();

    v8f d = wmma4x4(abb, bop);
#pragma unroll
    for (int v = 0; v < 8; ++v) stq[v][a * 16] = d[v];
    wave_sync();
  }

  // ---- Phase 4: atom gather + affine transform + mask --------------------
  // 4 residues * 14 atoms = 56 tasks across 32 lanes.
  for (int t = lane; t < 4 * 14; t += 32) {
    const int ri   = t / 14;
    const int atom = t - ri * 14;
    const int res  = resBase + ri;
    const int sq   = seq[res];
    const int g    = gidx[sq * 14 + atom];
    const float* G = &fr[(ri * 8 + g) * 16];
    const float* lp = litp + (sq * 14 + atom) * 3;
    const float l0 = lp[0], l1 = lp[1], l2 = lp[2];
    const float mk = amask[sq * 14 + atom];
    float* op = out + (res * 14 + atom) * 3;
#pragma unroll
    for (int r = 0; r < 3; ++r) {
      op[r] = (G[r * 4 + 0] * l0 + G[r * 4 + 1] * l1 + G[r * 4 + 2] * l2 +
               G[r * 4 + 3]) * mk;
    }
  }
}

extern "C" void kernel_launch(void* const* d_in, const int* in_sizes, int n_in,
                              void* d_out, int out_size, void* d_ws,
                              size_t ws_size, hipStream_t stream) {
  const float* dih     = (const float*)d_in[0];  // dihedrals
  const float* bbc     = (const float*)d_in[1];  // bb_coords
  const int*   seq     = (const int*)d_in[2];    // sequence
  const float* dframes = (const float*)d_in[3];  // default_frames
  const int*   gidx    = (const int*)d_in[4];    // group_idx
  const float* amask   = (const float*)d_in[5];  // atom_mask14
  const float* litp    = (const float*)d_in[6];  // lit_positions
  float*       out     = (float*)d_out;

  const int blocks = RES_TOTAL / RES_PER_BLK;  // 4096
  rotamer_projection_kernel<<<blocks, 256, 0, stream>>>(
      dih, bbc, seq, dframes, gidx, amask, litp, out);
}